// GSTokenizer_91104846283030
// MI455X (gfx1250) — compile-verified
//
#include <hip/hip_runtime.h>

typedef float v2f __attribute__((ext_vector_type(2)));
typedef float v8f __attribute__((ext_vector_type(8)));

#define GRID_RES 64
#define IMG      512
#define CHANNELS 64
#define HW       (IMG * IMG)

__device__ __forceinline__ int readlane_i(int v, int l) {
    return __builtin_amdgcn_readlane(v, l);
}
__device__ __forceinline__ float readlane_f(float v, int l) {
    union { float f; int i; } u;
    u.f = v;
    u.i = __builtin_amdgcn_readlane(u.i, l);
    return u.f;
}

__global__ __launch_bounds__(256) void gs_tokenizer_kernel(
    const int*   __restrict__ coords,   // (N,4) int32: b,i,j,k
    const float* __restrict__ feat,     // (4,64,512,512) f32
    const float* __restrict__ Km,       // (3,3) f32
    const float* __restrict__ Pm,       // (4,4) f32
    float*       __restrict__ out,      // N*64 features then N depth
    int Npts)
{
    const int lane = threadIdx.x & 31;
    const int wave = threadIdx.x >> 5;
    const long pbase = ((long)blockIdx.x * 8 + wave) * 16;  // 16 points / wave
    const int n = lane & 15;                                // this lane's point slot

    // ---- load point coords (half-waves duplicate; clamp for tail safety) ----
    long pidx = pbase + n;
    if (pidx >= Npts) pidx = (long)Npts - 1;
    const int4 c4 = ((const int4*)coords)[pidx];
    const int  bb = c4.x;
    const float s  = 2.0f / (float)(GRID_RES - 1);
    const float wx = (float)c4.y * s - 1.0f;
    const float wy = (float)c4.z * s - 1.0f;
    const float wz = (float)c4.w * s - 1.0f;

    // ---- A matrix (16x4): rows 0..3 = R (combined K*P projection), rest 0 ----
    // ISA A layout: VGPR0 holds K=0 (lanes0-15) / K=2 (lanes16-31); VGPR1: K=1 / K=3
    const int k0 = (lane < 16) ? 0 : 2;
    float a0 = 0.0f, a1 = 0.0f;
    if (n < 2) {
        // R[n][j] = sum_t K[n][t] * P[t][j]
        a0 = Km[n*3+0]*Pm[0*4+k0]   + Km[n*3+1]*Pm[1*4+k0]   + Km[n*3+2]*Pm[2*4+k0];
        a1 = Km[n*3+0]*Pm[0*4+k0+1] + Km[n*3+1]*Pm[1*4+k0+1] + Km[n*3+2]*Pm[2*4+k0+1];
    } else if (n == 2 || n == 3) {
        a0 = Pm[n*4+k0];
        a1 = Pm[n*4+k0+1];
    }
    v2f A; A.x = a0; A.y = a1;

    // ---- B matrix (4x16): column n = pts_h of point n; rows split (0,2)/(1,3) ----
    v2f B;
    B.x = (lane < 16) ? wx : wz;
    B.y = (lane < 16) ? wy : 1.0f;

    // ---- D = A x B : lane n<16 gets q0..q3 of point n in acc[0..3] ----
    v8f acc = {};
    acc = __builtin_amdgcn_wmma_f32_16x16x4_f32(
        /*neg_a=*/false, A, /*neg_b=*/false, B,
        /*c_mod=*/(short)0, acc, /*reuse_a=*/false, /*reuse_b=*/false);

    const float q0 = acc[0];   // u_h = K0 . cam  (w cancels in the x/y ratios)
    const float q1 = acc[1];   // v_h = K1 . cam
    const float q2 = acc[2];   // cam_z
    const float q3 = acc[3];   // cam_w

    const float xr = q0 / q2;
    const float yr = q1 / q2;
    const float depth = q2 / q3;

    const float if_ = floorf(yr);
    const float jf_ = floorf(xr);
    const float di = yr - if_;
    const float dj = xr - jf_;
    int i0 = (int)if_;
    int j0 = (int)jf_;
    int i1 = i0 + 1, j1 = j0 + 1;
    i0 = min(max(i0, 0), IMG - 1);
    i1 = min(max(i1, 0), IMG - 1);
    j0 = min(max(j0, 0), IMG - 1);
    j1 = min(max(j1, 0), IMG - 1);

    // depth output lives after the N*64 features
    if (lane < 16 && (pbase + n) < (long)Npts) {
        out[(size_t)Npts * 64 + (size_t)(pbase + n)] = depth;
    }

    // per-lane precompute; broadcast to SGPRs via readlane in the loop
    const int baseoff = bb * (CHANNELS * HW);   // batch base (elements)
    const int ro0 = i0 * IMG;
    const int ro1 = i1 * IMG;
    const int c0  = lane * 2;                   // this lane's channel pair

    #pragma unroll
    for (int t = 0; t < 16; ++t) {
        const long pt = pbase + t;
        // constant source lane -> v_readlane into SGPRs (uniform scalars)
        const int   tb  = readlane_i(baseoff, t);
        const int   tr0 = readlane_i(ro0,     t);
        const int   tr1 = readlane_i(ro1,     t);
        const int   tj0 = readlane_i(j0,      t);
        const int   tj1 = readlane_i(j1,      t);
        const float tdi = readlane_f(di,      t);
        const float tdj = readlane_f(dj,      t);

        // scalar part of every address is uniform; per-lane part is c0*HW
        const float* b0 = feat + (size_t)tb + (size_t)c0 * HW;
        const float* b1 = b0 + HW;

        const float f00a = b0[tr0 + tj0];
        const float f10a = b0[tr1 + tj0];
        const float f01a = b0[tr0 + tj1];
        const float f11a = b0[tr1 + tj1];
        const float f00b = b1[tr0 + tj0];
        const float f10b = b1[tr1 + tj0];
        const float f01b = b1[tr0 + tj1];
        const float f11b = b1[tr1 + tj1];

        const float omdi = 1.0f - tdi;
        const float omdj = 1.0f - tdj;
        const float ma  = omdi * f00a + tdi * f10a;
        const float mpa = omdi * f01a + tdi * f11a;
        const float mb  = omdi * f00b + tdi * f10b;
        const float mpb = omdi * f01b + tdi * f11b;

        float2 r;
        r.x = omdj * ma + tdj * mpa;
        r.y = omdj * mb + tdj * mpb;

        if (pt < (long)Npts) {
            *(float2*)(out + (size_t)pt * 64 + c0) = r;   // wave: 256B coalesced
        }
    }
}

extern "C" void kernel_launch(void* const* d_in, const int* in_sizes, int n_in,
                              void* d_out, int out_size, void* d_ws, size_t ws_size,
                              hipStream_t stream) {
    const int*   coords = (const int*)d_in[0];
    const float* feat   = (const float*)d_in[1];
    const float* Km     = (const float*)d_in[2];
    const float* Pm     = (const float*)d_in[3];
    float* out = (float*)d_out;

    const int Npts = in_sizes[0] / 4;
    const int pts_per_block = 128;                 // 8 waves x 16 points
    const int nblocks = (Npts + pts_per_block - 1) / pts_per_block;
    gs_tokenizer_kernel<<<nblocks, 256, 0, stream>>>(coords, feat, Km, Pm, out, Npts);
}